// HebbianPlasticLayer_30863634989576
// MI455X (gfx1250) — compile-verified
//
#include <hip/hip_runtime.h>
#include <math.h>

#define B_DIM 4096
#define DIN   4096
#define DOUT  4096

#define BM 128
#define BN 128
#define BK 32
#define LDK 36   // padded row stride (floats): 144B rows -> 16B aligned, conflict-free b64 frag reads

typedef __attribute__((ext_vector_type(2))) float v2f;
typedef __attribute__((ext_vector_type(4))) float v4f;
typedef __attribute__((ext_vector_type(8))) float v8f;

// fast sigmoid: v_exp_f32 + v_rcp_f32 (1-ulp), no IEEE divide chain
__device__ __forceinline__ float fast_sigmoid(float a) {
  return __builtin_amdgcn_rcpf(1.0f + __expf(-a));
}
// fast tanh: tanh(y) = 1 - 2/(1+exp(2y)); saturates correctly for |y| large
__device__ __forceinline__ float fast_tanh(float y) {
  return 1.0f - 2.0f * __builtin_amdgcn_rcpf(1.0f + __expf(2.0f * y));
}

// ---------------------------------------------------------------------------
// Kernel 1: per-row inverse norm  inv[b] = 1 / (||x_b|| + 1e-8)
// ---------------------------------------------------------------------------
__global__ __launch_bounds__(256) void row_invnorm_kernel(const float* __restrict__ x,
                                                          float* __restrict__ inv) {
  __shared__ float red[256];
  const int row = blockIdx.x;
  const float* xr = x + (size_t)row * DIN;
  float s = 0.f;
  const int base = threadIdx.x * 4;            // 256 threads * 4 floats * 4 passes = 4096
#pragma unroll
  for (int t = 0; t < 4; ++t) {
    v4f v = *(const v4f*)(xr + base + t * 1024);
    s += v.x * v.x + v.y * v.y + v.z * v.z + v.w * v.w;
  }
  red[threadIdx.x] = s;
  __syncthreads();
  for (int off = 128; off > 0; off >>= 1) {
    if (threadIdx.x < off) red[threadIdx.x] += red[threadIdx.x + off];
    __syncthreads();
  }
  if (threadIdx.x == 0) inv[row] = 1.0f / (sqrtf(red[0]) + 1e-8f);
}

// ---------------------------------------------------------------------------
// Kernel 2: fused dual-GEMM with f32 WMMA, software-pipelined tile loads.
//   Y[m,n] = sum_k x[m,k] * W[n,k]
//   S[m,n] = sum_k (x[m,k]^2 * inv[m]) * sigmoid(alpha[n,k])
//   out    = Y + bias + eta * tanh(Y + bias) * S
// ---------------------------------------------------------------------------
__global__ __launch_bounds__(256) void hebbian_fused_kernel(
    const float* __restrict__ x, const float* __restrict__ W,
    const float* __restrict__ alpha, const float* __restrict__ eta_p,
    const float* __restrict__ bias, const float* __restrict__ inv,
    float* __restrict__ out) {
  __shared__ float xs [BM][LDK];  // x tile               (rows = batch)
  __shared__ float wsL[BN][LDK];  // W tile               (rows = out-feature)
  __shared__ float asL[BN][LDK];  // sigmoid(alpha) tile

  const int tid  = threadIdx.x;
  const int lane = tid & 31;
  const int l16  = lane & 15;
  const int kh   = lane >> 4;               // K-half selector per WMMA layout
  const int waveId = tid >> 5;
  const int wm = (waveId >> 2) * 64;        // wave M origin: 0 / 64
  const int wn = (waveId & 3) * 32;         // wave N origin: 0 / 32 / 64 / 96

  const int mBlock = blockIdx.y * BM;
  const int nBlock = blockIdx.x * BN;

  // cooperative global->LDS mapping: 256 threads * float4 = 32 rows/pass, 4 passes
  const int ldRow = tid >> 3;               // 0..31
  const int ldCol = (tid & 7) * 4;          // 0..28

  // per-thread row base pointers for the staged loads
  const float* xp[4];
  const float* wp[4];
  const float* ap[4];
#pragma unroll
  for (int p = 0; p < 4; ++p) {
    const int r = ldRow + p * 32;
    xp[p] = x     + (size_t)(mBlock + r) * DIN + ldCol;
    wp[p] = W     + (size_t)(nBlock + r) * DIN + ldCol;
    ap[p] = alpha + (size_t)(nBlock + r) * DIN + ldCol;
  }

  const float eta = *eta_p;

  v8f zero = {};
  v8f accY[4][2], accS[4][2];
#pragma unroll
  for (int i = 0; i < 4; ++i)
#pragma unroll
    for (int j = 0; j < 2; ++j) { accY[i][j] = zero; accS[i][j] = zero; }

  // per-lane row inverse norms (same for both K-halves of the wave)
  float invA[4];
#pragma unroll
  for (int i = 0; i < 4; ++i) invA[i] = inv[mBlock + wm + 16 * i + l16];

  // ---- software pipeline: prologue load of tile kt=0 into registers ----
  v4f rx[4], rw[4], ra[4];
#pragma unroll
  for (int p = 0; p < 4; ++p) {
    rx[p] = *(const v4f*)(xp[p]);
    rw[p] = *(const v4f*)(wp[p]);
    ra[p] = *(const v4f*)(ap[p]);
  }

  for (int kt = 0; kt < DIN; kt += BK) {
    __syncthreads();          // previous compute done reading LDS
    // commit staged registers -> LDS (sigmoid applied to alpha here)
#pragma unroll
    for (int p = 0; p < 4; ++p) {
      const int r = ldRow + p * 32;
      *(v4f*)&xs [r][ldCol] = rx[p];
      *(v4f*)&wsL[r][ldCol] = rw[p];
      v4f sg;
      sg.x = fast_sigmoid(ra[p].x);
      sg.y = fast_sigmoid(ra[p].y);
      sg.z = fast_sigmoid(ra[p].z);
      sg.w = fast_sigmoid(ra[p].w);
      *(v4f*)&asL[r][ldCol] = sg;
    }
    __syncthreads();

    // issue next tile's global loads now; latency hidden by the WMMA loop below
    const int ktn = kt + BK;
    if (ktn < DIN) {
      if (ktn + BK < DIN) {   // prefetch two tiles ahead into cache (global_prefetch_b8)
        __builtin_prefetch(xp[0] + (size_t)(ktn + BK), 0, 3);
        __builtin_prefetch(wp[0] + (size_t)(ktn + BK), 0, 3);
        __builtin_prefetch(ap[0] + (size_t)(ktn + BK), 0, 3);
      }
#pragma unroll
      for (int p = 0; p < 4; ++p) {
        rx[p] = *(const v4f*)(xp[p] + ktn);
        rw[p] = *(const v4f*)(wp[p] + ktn);
        ra[p] = *(const v4f*)(ap[p] + ktn);
      }
    }

    // ---- WMMA compute on current LDS tile: 8 k-steps x 16 WMMAs ----
#pragma unroll
    for (int kk = 0; kk < BK; kk += 4) {
      v2f ax[4], a2[4], bw[2], ba[2];
#pragma unroll
      for (int i = 0; i < 4; ++i) {
        ax[i] = *(const v2f*)&xs[wm + 16 * i + l16][kk + 2 * kh];
        a2[i].x = ax[i].x * ax[i].x * invA[i];
        a2[i].y = ax[i].y * ax[i].y * invA[i];
      }
#pragma unroll
      for (int j = 0; j < 2; ++j) {
        bw[j] = *(const v2f*)&wsL[wn + 16 * j + l16][kk + 2 * kh];
        ba[j] = *(const v2f*)&asL[wn + 16 * j + l16][kk + 2 * kh];
      }
#pragma unroll
      for (int i = 0; i < 4; ++i)
#pragma unroll
        for (int j = 0; j < 2; ++j) {
          accY[i][j] = __builtin_amdgcn_wmma_f32_16x16x4_f32(
              false, ax[i], false, bw[j], (short)0, accY[i][j], false, false);
          accS[i][j] = __builtin_amdgcn_wmma_f32_16x16x4_f32(
              false, a2[i], false, ba[j], (short)0, accS[i][j], false, false);
        }
    }
  }

  // ---- epilogue: bias, fast tanh, eta, store ----
#pragma unroll
  for (int j = 0; j < 2; ++j) {
    const int n  = nBlock + wn + 16 * j + l16;
    const float bn = bias[n];
#pragma unroll
    for (int i = 0; i < 4; ++i) {
      const int mb = mBlock + wm + 16 * i + kh * 8;   // C layout: lanes 16-31 -> M+8
#pragma unroll
      for (int r = 0; r < 8; ++r) {
        const float yv = accY[i][j][r] + bn;
        const float sv = accS[i][j][r];
        out[(size_t)(mb + r) * DOUT + n] = yv + eta * fast_tanh(yv) * sv;
      }
    }
  }
}

// ---------------------------------------------------------------------------
// Launch
// ---------------------------------------------------------------------------
extern "C" void kernel_launch(void* const* d_in, const int* in_sizes, int n_in,
                              void* d_out, int out_size, void* d_ws, size_t ws_size,
                              hipStream_t stream) {
  const float* x     = (const float*)d_in[0];
  const float* W     = (const float*)d_in[1];
  const float* alpha = (const float*)d_in[2];
  const float* eta   = (const float*)d_in[3];
  const float* bias  = (const float*)d_in[4];
  float* out = (float*)d_out;
  float* inv = (float*)d_ws;   // 4096 floats of scratch

  row_invnorm_kernel<<<B_DIM, 256, 0, stream>>>(x, inv);

  dim3 grid(DOUT / BN, B_DIM / BM);   // 32 x 32 workgroups
  hebbian_fused_kernel<<<grid, 256, 0, stream>>>(x, W, alpha, eta, bias, inv, out);
}